// PointNetSetAbstractionMsg_31980326486608
// MI455X (gfx1250) — compile-verified
//
#include <hip/hip_runtime.h>
#include <hip/hip_bf16.h>
#include <stdint.h>

typedef __attribute__((ext_vector_type(2))) float v2f;
typedef __attribute__((ext_vector_type(8))) float v8f;

#define FPS_THREADS 1024

// ---------------------------------------------------------------------------
// Farthest point sampling: one block per batch. All N=8192 points cached in
// registers (8 per thread). Per step: distance update + wave32 shfl arg-max,
// then a 32-wave LDS arg-max (2 barriers per step instead of a 10-level tree).
// ---------------------------------------------------------------------------
__global__ void __launch_bounds__(FPS_THREADS) fps_kernel(
    const float* __restrict__ xyz, int* __restrict__ fps_idx, int N, int S) {
  const int b = blockIdx.x;
  const int t = threadIdx.x;
  const int lane = t & 31;
  const int wave = t >> 5;
  const float* X = xyz + (size_t)b * N * 3;
  const int PPT = N / FPS_THREADS;  // 8
  float px[8], py[8], pz[8], dist[8];
  for (int i = 0; i < PPT; ++i) {
    int n = t + i * FPS_THREADS;
    px[i] = X[n * 3 + 0];
    py[i] = X[n * 3 + 1];
    pz[i] = X[n * 3 + 2];
    dist[i] = 1e10f;
  }
  __shared__ float swv[32];
  __shared__ int swi[32];
  __shared__ float csh[3];
  __shared__ int farsh;
  int far = 0;
  for (int s = 0; s < S; ++s) {
    if (t == 0) {
      fps_idx[(size_t)b * S + s] = far;
      csh[0] = X[far * 3 + 0];
      csh[1] = X[far * 3 + 1];
      csh[2] = X[far * 3 + 2];
    }
    __syncthreads();
    float cx = csh[0], cy = csh[1], cz = csh[2];
    float best = -1.0f;
    int bi = 0x7fffffff;
    for (int i = 0; i < PPT; ++i) {
      float dx = px[i] - cx, dy = py[i] - cy, dz = pz[i] - cz;
      float d = dx * dx + dy * dy + dz * dz;
      float nd = fminf(dist[i], d);
      dist[i] = nd;
      int n = t + i * FPS_THREADS;
      if (nd > best || (nd == best && n < bi)) { best = nd; bi = n; }
    }
    // wave32 arg-max reduction (first-index tie break, matching jnp.argmax)
    for (int off = 16; off > 0; off >>= 1) {
      float v2 = __shfl_xor(best, off, 32);
      int i2 = __shfl_xor(bi, off, 32);
      if (v2 > best || (v2 == best && i2 < bi)) { best = v2; bi = i2; }
    }
    if (lane == 0) { swv[wave] = best; swi[wave] = bi; }
    __syncthreads();
    if (wave == 0) {
      float v = swv[lane];
      int idx = swi[lane];
      for (int off = 16; off > 0; off >>= 1) {
        float v2 = __shfl_xor(v, off, 32);
        int i2 = __shfl_xor(idx, off, 32);
        if (v2 > v || (v2 == v && i2 < idx)) { v = v2; idx = i2; }
      }
      if (lane == 0) farsh = idx;
    }
    __syncthreads();
    far = farsh;
  }
}

// ---------------------------------------------------------------------------
// Gather centroid coordinates; also emit transposed (B,3,S) output directly.
// ---------------------------------------------------------------------------
__global__ void gather_centroids(const float* __restrict__ xyz,
                                 const int* __restrict__ fps_idx,
                                 float* __restrict__ new_xyz,
                                 float* __restrict__ out_xyz, int B, int N,
                                 int S) {
  int i = blockIdx.x * blockDim.x + threadIdx.x;
  if (i >= B * S) return;
  int b = i / S, s = i % S;
  int n = fps_idx[i];
  for (int d = 0; d < 3; ++d) {
    float v = xyz[((size_t)b * N + n) * 3 + d];
    new_xyz[(size_t)i * 3 + d] = v;
    out_xyz[((size_t)b * 3 + d) * S + s] = v;
  }
}

// ---------------------------------------------------------------------------
// Ball query: one wave32 per centroid. Ballot + prefix popcount keeps the
// first K in-index-order neighbors (exact reference semantics), pad with the
// first neighbor (centroid itself is always within radius -> cnt >= 1).
// ---------------------------------------------------------------------------
__global__ void query_ball_kernel(const float* __restrict__ xyz,
                                  const float* __restrict__ new_xyz,
                                  int* __restrict__ gidx, int B, int N, int S,
                                  int K, float r2) {
  int wid = (blockIdx.x * blockDim.x + threadIdx.x) >> 5;
  int lane = threadIdx.x & 31;
  if (wid >= B * S) return;
  int b = wid / S;
  const float* X = xyz + (size_t)b * N * 3;
  float cx = new_xyz[(size_t)wid * 3 + 0];
  float cy = new_xyz[(size_t)wid * 3 + 1];
  float cz = new_xyz[(size_t)wid * 3 + 2];
  int* out = gidx + (size_t)wid * K;
  int cnt = 0;
  int first = -1;
  for (int j0 = 0; j0 < N && cnt < K; j0 += 32) {
    int j = j0 + lane;
    float dx = X[j * 3 + 0] - cx;
    float dy = X[j * 3 + 1] - cy;
    float dz = X[j * 3 + 2] - cz;
    float d2 = dx * dx + dy * dy + dz * dz;
    bool in = (d2 <= r2);
    unsigned mask = (unsigned)__ballot(in);
    int pos = cnt + __popc(mask & ((1u << lane) - 1u));
    if (in && pos < K) out[pos] = j;
    if (first < 0 && mask != 0u) first = j0 + __ffs((int)mask) - 1;
    cnt += __popc(mask);
  }
  if (cnt < K) {
    int f = (first >= 0) ? first : 0;
    for (int p = cnt + lane; p < K; p += 32) out[p] = f;
  }
}

// ---------------------------------------------------------------------------
// Gather grouped features (B,S,K,16) as float4s.
// ---------------------------------------------------------------------------
__global__ void group_gather(const float* __restrict__ feat,
                             const int* __restrict__ gidx,
                             float* __restrict__ x0, int B, int N, size_t rows) {
  size_t i = blockIdx.x * (size_t)blockDim.x + threadIdx.x;
  size_t total = rows * 4;
  if (i >= total) return;
  size_t row = i >> 2;
  int q = (int)(i & 3);
  size_t perB = rows / (size_t)B;
  int b = (int)(row / perB);
  int n = gidx[row];
  const float4* src = (const float4*)(feat + ((size_t)b * N + n) * 16);
  float4* dst = (float4*)(x0 + row * 16);
  dst[q] = src[q];
}

// ---------------------------------------------------------------------------
// GEMM: y[row, co] = sum_ci x[row,ci] * w[co,ci] + bias[co]
// Templated on CIN so the K-loop fully unrolls and the B (weight) fragments
// stay register-resident for the wave's lifetime. Each wave owns one 16-col
// tile and a strip of M tiles: B frags + bias loaded once, then per M tile
// only A loads + V_WMMA_F32_16X16X4_F32 + store.
// A frag (16x4 f32): lanes 0-15 -> K=0,1 ; lanes 16-31 -> K=2,3 (ISA layout).
// B frag (4x16 f32): lane n holds w[co=n][k+0..1] (lanes<16) / k+2..3 (>=16).
// Early-exit is wave-uniform so EXEC is all-1s at every WMMA.
// ---------------------------------------------------------------------------
#define M_TILES_PER_WAVE 8

template <int CIN>
__global__ void __launch_bounds__(256) gemm_bias_wmma(
    const float* __restrict__ x, const float* __restrict__ w,
    const float* __restrict__ bias, float* __restrict__ y, int rows,
    int Cout) {
  constexpr int KF = CIN / 4;
  int waveId = blockIdx.x * 8 + (threadIdx.x >> 5);
  int lane = threadIdx.x & 31;
  int tilesN = Cout >> 4;
  int tilesM = rows >> 4;
  int tn = waveId % tilesN;
  int mStart = (waveId / tilesN) * M_TILES_PER_WAVE;
  if (mStart >= tilesM) return;  // uniform per wave
  int lrow = lane & 15;
  int koff = (lane < 16) ? 0 : 2;

  // Preload all B fragments (weights) for this column tile into registers.
  const float* wr = w + (size_t)((tn << 4) + lrow) * CIN + koff;
  v2f bf[KF];
#pragma unroll
  for (int k = 0; k < KF; ++k) bf[k] = v2f{wr[4 * k], wr[4 * k + 1]};
  float bb = bias[(tn << 4) + lrow];

  int mEnd = mStart + M_TILES_PER_WAVE;
  if (mEnd > tilesM) mEnd = tilesM;
  for (int tm = mStart; tm < mEnd; ++tm) {
    const float* xr = x + (size_t)((tm << 4) + lrow) * CIN + koff;
    if (tm + 1 < mEnd) {
      __builtin_prefetch(xr + 16 * CIN, 0, 3);  // global_prefetch_b8 next tile
    }
    v8f c = {};
#pragma unroll
    for (int k = 0; k < KF; ++k) {
      v2f a = {xr[4 * k], xr[4 * k + 1]};
      c = __builtin_amdgcn_wmma_f32_16x16x4_f32(false, a, false, bf[k],
                                                (short)0, c, false, false);
    }
    int orow = (tm << 4) + ((lane < 16) ? 0 : 8);
    float* yp = y + (size_t)orow * Cout + (tn << 4) + lrow;
#pragma unroll
    for (int v = 0; v < 8; ++v) yp[(size_t)v * Cout] = c[v] + bb;
  }
}

// ---------------------------------------------------------------------------
// Per-channel sum / sum-of-squares (for BN) via LDS atomics + final global add.
// ---------------------------------------------------------------------------
__global__ void __launch_bounds__(256) reduce_stats(const float* __restrict__ y,
                                                    float* __restrict__ sum,
                                                    float* __restrict__ sumsq,
                                                    size_t rows, int Cout) {
  __shared__ float ssum[128];
  __shared__ float ssq[128];
  for (int cch = threadIdx.x; cch < Cout; cch += blockDim.x) {
    ssum[cch] = 0.0f;
    ssq[cch] = 0.0f;
  }
  __syncthreads();
  size_t total = rows * (size_t)Cout;
  size_t stride = (size_t)gridDim.x * blockDim.x;
  for (size_t i = blockIdx.x * (size_t)blockDim.x + threadIdx.x; i < total;
       i += stride) {
    float v = y[i];
    int cch = (int)(i % (size_t)Cout);
    atomicAdd(&ssum[cch], v);
    atomicAdd(&ssq[cch], v * v);
  }
  __syncthreads();
  for (int cch = threadIdx.x; cch < Cout; cch += blockDim.x) {
    atomicAdd(&sum[cch], ssum[cch]);
    atomicAdd(&sumsq[cch], ssq[cch]);
  }
}

__global__ void finalize_stats(const float* __restrict__ sum,
                               const float* __restrict__ sumsq,
                               const float* __restrict__ gamma,
                               const float* __restrict__ beta,
                               float* __restrict__ scale,
                               float* __restrict__ shift, float cntInv,
                               int Cout) {
  int cch = threadIdx.x;
  if (cch >= Cout) return;
  float mean = sum[cch] * cntInv;
  float var = sumsq[cch] * cntInv - mean * mean;
  float sc = gamma[cch] * rsqrtf(var + 1e-5f);
  scale[cch] = sc;
  shift[cch] = beta[cch] - mean * sc;
}

__global__ void __launch_bounds__(256) bn_relu(float* __restrict__ y,
                                               const float* __restrict__ scale,
                                               const float* __restrict__ shift,
                                               size_t rows, int Cout) {
  size_t total = rows * (size_t)Cout;
  size_t stride = (size_t)gridDim.x * blockDim.x;
  for (size_t i = blockIdx.x * (size_t)blockDim.x + threadIdx.x; i < total;
       i += stride) {
    int cch = (int)(i % (size_t)Cout);
    float v = y[i] * scale[cch] + shift[cch];
    y[i] = v > 0.0f ? v : 0.0f;
  }
}

// ---------------------------------------------------------------------------
// Max over K, writing transposed (B, Ctot, S) slice at channel offset coff.
// ---------------------------------------------------------------------------
__global__ void max_pool_k(const float* __restrict__ y, float* __restrict__ out,
                           int B, int S, int K, int Cout, int coff, int Ctot) {
  int i = blockIdx.x * blockDim.x + threadIdx.x;
  if (i >= B * S * Cout) return;
  int cch = i % Cout;
  int bs = i / Cout;
  int b = bs / S, s = bs % S;
  const float* yp = y + ((size_t)bs * K) * Cout + cch;
  float m = yp[0];
  for (int k = 1; k < K; ++k) m = fmaxf(m, yp[(size_t)k * Cout]);
  out[((size_t)b * Ctot + coff + cch) * S + s] = m;
}

static inline size_t alignUp(size_t x, size_t a) { return (x + a - 1) / a * a; }
static inline unsigned ceilDiv(size_t a, size_t b) {
  return (unsigned)((a + b - 1) / b);
}

static void launch_gemm(const float* x, const float* w, const float* bias,
                        float* y, size_t rows, int Cin, int Cout,
                        hipStream_t stream) {
  size_t tilesM = rows >> 4;
  size_t tilesN = (size_t)Cout >> 4;
  size_t waves = tilesN * ((tilesM + M_TILES_PER_WAVE - 1) / M_TILES_PER_WAVE);
  unsigned blocks = ceilDiv(waves, 8);
  switch (Cin) {
    case 16:
      gemm_bias_wmma<16><<<blocks, 256, 0, stream>>>(x, w, bias, y, (int)rows,
                                                     Cout);
      break;
    case 32:
      gemm_bias_wmma<32><<<blocks, 256, 0, stream>>>(x, w, bias, y, (int)rows,
                                                     Cout);
      break;
    case 64:
      gemm_bias_wmma<64><<<blocks, 256, 0, stream>>>(x, w, bias, y, (int)rows,
                                                     Cout);
      break;
    default:
      gemm_bias_wmma<96><<<blocks, 256, 0, stream>>>(x, w, bias, y, (int)rows,
                                                     Cout);
      break;
  }
}

extern "C" void kernel_launch(void* const* d_in, const int* in_sizes, int n_in,
                              void* d_out, int out_size, void* d_ws,
                              size_t ws_size, hipStream_t stream) {
  (void)in_sizes; (void)n_in; (void)out_size; (void)ws_size;
  const int B = 4, N = 8192, S = 2048;
  const float radii[3] = {0.1f, 0.2f, 0.4f};
  const int Ks[3] = {16, 32, 64};
  const int mlp[3][4] = {{16, 32, 32, 64}, {16, 64, 64, 128}, {16, 64, 96, 128}};
  const int Ctot = 320;

  const float* xyz = (const float*)d_in[0];
  const float* feat = (const float*)d_in[1];

  // workspace carve-out
  uint8_t* p = (uint8_t*)d_ws;
  int* fpsIdx = (int*)p;      p += alignUp((size_t)B * S * sizeof(int), 256);
  float* newXyz = (float*)p;  p += alignUp((size_t)B * S * 3 * sizeof(float), 256);
  int* gidx = (int*)p;        p += alignUp((size_t)B * S * 64 * sizeof(int), 256);
  float* sumBuf = (float*)p;  p += alignUp(128 * sizeof(float), 256);
  float* sqBuf = (float*)p;   p += alignUp(128 * sizeof(float), 256);
  float* scaleBuf = (float*)p; p += alignUp(128 * sizeof(float), 256);
  float* shiftBuf = (float*)p; p += alignUp(128 * sizeof(float), 256);
  const size_t maxRows = (size_t)B * S * 64;
  float* xbuf = (float*)p;    p += alignUp(maxRows * 96 * sizeof(float), 256);
  float* ybuf = (float*)p;    // maxRows * 128 floats

  float* outXyz = (float*)d_out;                       // (B,3,S)
  float* outPts = (float*)d_out + (size_t)B * 3 * S;   // (B,320,S)

  // 1) FPS + centroid gather (writes transposed new_xyz to d_out directly)
  fps_kernel<<<B, FPS_THREADS, 0, stream>>>(xyz, fpsIdx, N, S);
  gather_centroids<<<ceilDiv((size_t)B * S, 256), 256, 0, stream>>>(
      xyz, fpsIdx, newXyz, outXyz, B, N, S);

  int coff = 0;
  for (int sc = 0; sc < 3; ++sc) {
    const int K = Ks[sc];
    const size_t rows = (size_t)B * S * K;
    const float r2 = radii[sc] * radii[sc];

    // 2) ball query (one wave per centroid)
    query_ball_kernel<<<ceilDiv((size_t)B * S * 32, 256), 256, 0, stream>>>(
        xyz, newXyz, gidx, B, N, S, K, r2);

    // 3) gather grouped features into xbuf (rows x 16)
    group_gather<<<ceilDiv(rows * 4, 256), 256, 0, stream>>>(feat, gidx, xbuf,
                                                             B, N, rows);

    // 4) MLP: 3 x (WMMA gemm + BN-stats + finalize + bn_relu), ping-pong
    const float* cur = xbuf;
    for (int l = 0; l < 3; ++l) {
      const int Cin = mlp[sc][l];
      const int Cout = mlp[sc][l + 1];
      float* nxt = (l & 1) ? xbuf : ybuf;
      const float* W = (const float*)d_in[2 + sc * 12 + l * 4 + 0];
      const float* Bi = (const float*)d_in[2 + sc * 12 + l * 4 + 1];
      const float* Ga = (const float*)d_in[2 + sc * 12 + l * 4 + 2];
      const float* Be = (const float*)d_in[2 + sc * 12 + l * 4 + 3];

      launch_gemm(cur, W, Bi, nxt, rows, Cin, Cout, stream);

      hipMemsetAsync(sumBuf, 0, Cout * sizeof(float), stream);
      hipMemsetAsync(sqBuf, 0, Cout * sizeof(float), stream);
      reduce_stats<<<2048, 256, 0, stream>>>(nxt, sumBuf, sqBuf, rows, Cout);
      finalize_stats<<<1, 128, 0, stream>>>(sumBuf, sqBuf, Ga, Be, scaleBuf,
                                            shiftBuf, 1.0f / (float)rows, Cout);
      bn_relu<<<2048, 256, 0, stream>>>(nxt, scaleBuf, shiftBuf, rows, Cout);
      cur = nxt;
    }

    // 5) max over K -> transposed (B, Ctot, S) slice
    const int CoutF = mlp[sc][3];
    max_pool_k<<<ceilDiv((size_t)B * S * CoutF, 256), 256, 0, stream>>>(
        cur, outPts, B, S, K, CoutF, coff, Ctot);
    coff += CoutF;
  }
}